// SkipGCN_67164698575458
// MI455X (gfx1250) — compile-verified
//
#include <hip/hip_runtime.h>
#include <hip/hip_bf16.h>

typedef float v2f __attribute__((ext_vector_type(2)));
typedef float v8f __attribute__((ext_vector_type(8)));

#define N_NODES 100000
#define FDIM 128

// ---------------- utility: zero / init ----------------
__global__ void zero_f32_kernel(float* __restrict__ p, size_t n) {
    size_t i = (size_t)blockIdx.x * blockDim.x + threadIdx.x;
    size_t stride = (size_t)gridDim.x * blockDim.x;
    for (; i < n; i += stride) p[i] = 0.0f;
}

__global__ void fill_f32_kernel(float* __restrict__ p, float v, int n) {
    int i = blockIdx.x * blockDim.x + threadIdx.x;
    if (i < n) p[i] = v;
}

// ---------------- degree + normalization ----------------
__global__ void deg_accum_kernel(const int* __restrict__ dst, int E,
                                 float* __restrict__ deg) {
    int e = blockIdx.x * blockDim.x + threadIdx.x;
    if (e < E) atomicAdd(&deg[dst[e]], 1.0f);
}

__global__ void dinv_kernel(float* __restrict__ deg, int n) {
    int i = blockIdx.x * blockDim.x + threadIdx.x;
    if (i < n) {
        float d = deg[i];
        deg[i] = (d > 0.0f) ? rsqrtf(d) : 0.0f;  // in-place -> becomes dinv
    }
}

// ---------------- GEMM1: H1 = X @ W1 via V_WMMA_F32_16X16X4_F32 ----------------
// Wave-per-16x16-tile; 8 waves/block cover N=128 for a 16-row slab.
// A (16x4 f32) per-lane: lanes 0-15 hold K=k0+0,k0+1; lanes 16-31 hold k0+2,k0+3
// -> contiguous float2 load per lane.
// Two independent accumulator chains (K in [0,64) and [64,128)) to double the
// WMMA dependency distance and hide XDL latency.
__global__ __launch_bounds__(256) void gemm1_wmma_kernel(
    const float* __restrict__ X, const float* __restrict__ W,
    float* __restrict__ H, int M) {
    const int wave  = threadIdx.x >> 5;   // 0..7 -> column tile
    const int lane  = threadIdx.x & 31;
    const int row16 = lane & 15;
    const int khalf = lane >> 4;          // 0 or 1
    const int rowBase = blockIdx.x * 16;
    const int colBase = wave * 16;
    if (rowBase >= M) return;

    v8f c0 = {};
    v8f c1 = {};
    const float* xrow = X + (size_t)(rowBase + row16) * FDIM;
    const int HALF = FDIM / 2;            // 64

    #pragma unroll 4
    for (int k0 = 0; k0 < HALF; k0 += 4) {
        const int ka = k0 + 2 * khalf;
        const int kb = ka + HALF;
        v2f a0, b0, a1, b1;
        a0.x = xrow[ka];
        a0.y = xrow[ka + 1];
        a1.x = xrow[kb];
        a1.y = xrow[kb + 1];
        // B (4x16 f32): V0 = row ka (lanes = columns), V1 = row ka+1
        b0.x = W[(size_t)ka * FDIM + colBase + row16];
        b0.y = W[(size_t)(ka + 1) * FDIM + colBase + row16];
        b1.x = W[(size_t)kb * FDIM + colBase + row16];
        b1.y = W[(size_t)(kb + 1) * FDIM + colBase + row16];
        c0 = __builtin_amdgcn_wmma_f32_16x16x4_f32(
                false, a0, false, b0, (short)0, c0, false, false);
        c1 = __builtin_amdgcn_wmma_f32_16x16x4_f32(
                false, a1, false, b1, (short)0, c1, false, false);
    }
    v8f c = c0 + c1;

    // D layout: VGPR r -> M = r + 8*khalf, N = colBase + row16 (lane-coalesced)
    float* hbase = H + (size_t)(rowBase + 8 * khalf) * FDIM + colBase + row16;
    #pragma unroll
    for (int r = 0; r < 8; ++r)
        hbase[(size_t)r * FDIM] = c[r];
}

// ---------------- propagation layer 1: wave per edge, float4 per lane ----------
__global__ __launch_bounds__(256) void scatter1_kernel(
    const int* __restrict__ src, const int* __restrict__ dst, int E,
    const float* __restrict__ dinv, const float* __restrict__ H1,
    float* __restrict__ A1) {
    int e = blockIdx.x * 8 + (threadIdx.x >> 5);
    if (e >= E) return;
    const int lane = threadIdx.x & 31;
    const int s = src[e];
    const int d = dst[e];
    const float norm = dinv[s] * dinv[d];
    float4 v = ((const float4*)(H1 + (size_t)s * FDIM))[lane];
    float* ap = A1 + (size_t)d * FDIM + lane * 4;
    atomicAdd(ap + 0, v.x * norm);
    atomicAdd(ap + 1, v.y * norm);
    atomicAdd(ap + 2, v.z * norm);
    atomicAdd(ap + 3, v.w * norm);
}

// self-loop + bias + ReLU (after all scatter-adds)
__global__ void finalize1_kernel(float* __restrict__ A1,
                                 const float* __restrict__ H1,
                                 const float* __restrict__ dinv,
                                 const float* __restrict__ b1, size_t total) {
    size_t i = (size_t)blockIdx.x * blockDim.x + threadIdx.x;
    size_t stride = (size_t)gridDim.x * blockDim.x;
    for (; i < total; i += stride) {
        int node = (int)(i >> 7);
        int ch   = (int)(i & (FDIM - 1));
        float di = dinv[node];
        float v = A1[i] + H1[i] * di * di + b1[ch];
        A1[i] = v > 0.0f ? v : 0.0f;
    }
}

// ---------------- layer 2 + skip projections: wave per node (N=2) --------------
__global__ __launch_bounds__(256) void dense2_kernel(
    const float* __restrict__ A1, const float* __restrict__ X,
    const float* __restrict__ W2, const float* __restrict__ Wsk,
    float* __restrict__ H2, float* __restrict__ SK, int n) {
    int node = blockIdx.x * 8 + (threadIdx.x >> 5);
    if (node >= n) return;
    const int lane = threadIdx.x & 31;
    const int k = lane * 4;
    float4 a = ((const float4*)(A1 + (size_t)node * FDIM))[lane];
    float4 x = ((const float4*)(X  + (size_t)node * FDIM))[lane];

    float h0 = a.x * W2[(k + 0) * 2 + 0] + a.y * W2[(k + 1) * 2 + 0] +
               a.z * W2[(k + 2) * 2 + 0] + a.w * W2[(k + 3) * 2 + 0];
    float h1 = a.x * W2[(k + 0) * 2 + 1] + a.y * W2[(k + 1) * 2 + 1] +
               a.z * W2[(k + 2) * 2 + 1] + a.w * W2[(k + 3) * 2 + 1];
    float s0 = x.x * Wsk[(k + 0) * 2 + 0] + x.y * Wsk[(k + 1) * 2 + 0] +
               x.z * Wsk[(k + 2) * 2 + 0] + x.w * Wsk[(k + 3) * 2 + 0];
    float s1 = x.x * Wsk[(k + 0) * 2 + 1] + x.y * Wsk[(k + 1) * 2 + 1] +
               x.z * Wsk[(k + 2) * 2 + 1] + x.w * Wsk[(k + 3) * 2 + 1];

    #pragma unroll
    for (int off = 16; off > 0; off >>= 1) {
        h0 += __shfl_down(h0, off, 32);
        h1 += __shfl_down(h1, off, 32);
        s0 += __shfl_down(s0, off, 32);
        s1 += __shfl_down(s1, off, 32);
    }
    if (lane == 0) {
        H2[(size_t)node * 2 + 0] = h0;
        H2[(size_t)node * 2 + 1] = h1;
        SK[(size_t)node * 2 + 0] = s0;
        SK[(size_t)node * 2 + 1] = s1;
    }
}

// ---------------- propagation layer 2: thread per edge (2 channels) ------------
__global__ void scatter2_kernel(const int* __restrict__ src,
                                const int* __restrict__ dst, int E,
                                const float* __restrict__ dinv,
                                const float* __restrict__ H2,
                                float* __restrict__ OUT) {
    int e = blockIdx.x * blockDim.x + threadIdx.x;
    if (e >= E) return;
    const int s = src[e];
    const int d = dst[e];
    const float norm = dinv[s] * dinv[d];
    atomicAdd(&OUT[(size_t)d * 2 + 0], H2[(size_t)s * 2 + 0] * norm);
    atomicAdd(&OUT[(size_t)d * 2 + 1], H2[(size_t)s * 2 + 1] * norm);
}

// self-loop + biases + skip
__global__ void finalize2_kernel(float* __restrict__ OUT,
                                 const float* __restrict__ H2,
                                 const float* __restrict__ SK,
                                 const float* __restrict__ dinv,
                                 const float* __restrict__ b2,
                                 const float* __restrict__ bsk, int n) {
    int i = blockIdx.x * blockDim.x + threadIdx.x;
    if (i >= n) return;
    float di = dinv[i];
    float dd = di * di;
    OUT[(size_t)i * 2 + 0] += H2[(size_t)i * 2 + 0] * dd + b2[0] + SK[(size_t)i * 2 + 0] + bsk[0];
    OUT[(size_t)i * 2 + 1] += H2[(size_t)i * 2 + 1] * dd + b2[1] + SK[(size_t)i * 2 + 1] + bsk[1];
}

// ---------------- launch ----------------
extern "C" void kernel_launch(void* const* d_in, const int* in_sizes, int n_in,
                              void* d_out, int out_size, void* d_ws, size_t ws_size,
                              hipStream_t stream) {
    const float* x   = (const float*)d_in[0];
    const int*   ei  = (const int*)d_in[1];
    const float* W1  = (const float*)d_in[2];
    const float* b1  = (const float*)d_in[3];
    const float* W2  = (const float*)d_in[4];
    const float* b2  = (const float*)d_in[5];
    const float* Wsk = (const float*)d_in[6];
    const float* bsk = (const float*)d_in[7];
    float* out = (float*)d_out;

    const int N = in_sizes[0] / FDIM;     // 100000
    const int E = in_sizes[1] / 2;        // 1600000
    const int* src = ei;
    const int* dst = ei + E;

    // workspace carve-up (all offsets 16B aligned)
    char* ws = (char*)d_ws;
    float* dinv = (float*)ws;                              // N floats (deg -> dinv)
    float* H1   = (float*)(ws + (size_t)N * sizeof(float));          // N*128
    float* A1   = H1 + (size_t)N * FDIM;                             // N*128
    float* H2   = A1 + (size_t)N * FDIM;                             // N*2
    float* SK   = H2 + (size_t)N * 2;                                // N*2

    // 0) init: deg = 1.0 (self-loop), A1 = 0, OUT = 0
    fill_f32_kernel<<<(N + 255) / 256, 256, 0, stream>>>(dinv, 1.0f, N);
    zero_f32_kernel<<<2048, 256, 0, stream>>>(A1, (size_t)N * FDIM);
    zero_f32_kernel<<<512, 256, 0, stream>>>(out, (size_t)out_size);

    // 1) degree accumulation + dinv = rsqrt(deg)
    deg_accum_kernel<<<(E + 255) / 256, 256, 0, stream>>>(dst, E, dinv);
    dinv_kernel<<<(N + 255) / 256, 256, 0, stream>>>(dinv, N);

    // 2) H1 = X @ W1   (WMMA f32 16x16x4, wave-per-tile, dual accumulators)
    gemm1_wmma_kernel<<<(N + 15) / 16, 256, 0, stream>>>(x, W1, H1, N);

    // 3) edge propagation layer 1 (wave per edge) + self-loop/bias/ReLU
    scatter1_kernel<<<(E + 7) / 8, 256, 0, stream>>>(src, dst, E, dinv, H1, A1);
    finalize1_kernel<<<2048, 256, 0, stream>>>(A1, H1, dinv, b1, (size_t)N * FDIM);

    // 4) layer-2 + skip projections (wave per node, N=2 -> VALU dot + shuffle)
    dense2_kernel<<<(N + 7) / 8, 256, 0, stream>>>(A1, x, W2, Wsk, H2, SK, N);

    // 5) edge propagation layer 2 + finalize into OUT
    scatter2_kernel<<<(E + 255) / 256, 256, 0, stream>>>(src, dst, E, dinv, H2, out);
    finalize2_kernel<<<(N + 255) / 256, 256, 0, stream>>>(out, H2, SK, dinv, b2, bsk, N);
}